// ElasticDeformation_472446402678
// MI455X (gfx1250) — compile-verified
//
#include <hip/hip_runtime.h>

// Elastic deformation warp for (B=4, C=1, D=64, H=256, W=256), coarse flow 8^3 x 3.
// Flow upsampling (the reference's einsum chain) is done with V_WMMA_F32_16X16X4_F32
// per 16x16 (h,w) tile; the gather stage is plain bandwidth-bound VMEM.

typedef __attribute__((ext_vector_type(2))) float v2f;
typedef __attribute__((ext_vector_type(8))) float v8f;

__device__ __forceinline__ float lerpf(float a, float b, float f) { return fmaf(f, b - a, a); }

// reflection about pixel centers 0 and n-1 (align_corners=True); m = 2*(n-1)
__device__ __forceinline__ float reflectf(float v, float m, float invm) {
    float t = fabsf(v);
    t = fmaf(-m, floorf(t * invm), t);   // t mod m  (t >= 0)
    return fminf(t, m - t);
}

__global__ __launch_bounds__(256) void elastic_warp_kernel(
        const float* __restrict__ x,      // (4,1,64,256,256)
        const float* __restrict__ fc,     // (4,3,8,8,8) channels [z,y,x]-scaled
        float* __restrict__ out)          // (4,1,64,256,256)
{
    __shared__ float Gs[3][8][8];         // coarse flow with D-axis collapsed at this d

    const int tid  = threadIdx.x;
    const int wave = tid >> 5;
    const int lane = tid & 31;
    const int lh   = lane >> 4;           // lane half (0/1)
    const int col  = lane & 15;

    // block decode: grid.x = B * D * (H/16) * (W/128) = 4*64*16*2 = 8192
    const int wseg  =  blockIdx.x        & 1;
    const int htile = (blockIdx.x >> 1)  & 15;
    const int d     = (blockIdx.x >> 5)  & 63;
    const int b     =  blockIdx.x >> 11;

    const int h0 = htile << 4;
    const int w0 = (wseg << 7) + (wave << 4);

    // ---------- Stage 1: G[c][j][k] = lerp over coarse D-axis at this d ----------
    const float pd = (float)d * (7.0f / 63.0f);
    int i0 = (int)pd; if (i0 > 6) i0 = 6;
    const float fd = pd - (float)i0;
    if (tid < 192) {
        const int c = tid >> 6, j = (tid >> 3) & 7, k = tid & 7;
        const float* p = fc + ((((b * 3 + c) * 8 + i0) * 8 + j) * 8 + k);
        Gs[c][j][k] = lerpf(p[0], p[64], fd);   // stride between i-slices = 8*8
    }
    __syncthreads();

    // ---------- Stage 2: build WMMA operands; F(16x16) = WH_tile(16x8) * T(8x16) ----------
    // A layout (fp32 16x4 per chunk): lane L, VGPR v holds A[L%16][4*kc + 2*(L/16) + v]
    const int hA = h0 + col;
    const float ph = (float)hA * (7.0f / 255.0f);
    int j0 = (int)ph; if (j0 > 6) j0 = 6;
    const float fh = ph - (float)j0;

    v2f a0, a1;
    {
        const int ja = lh << 1;
        float av[4];
#pragma unroll
        for (int kc = 0; kc < 2; ++kc)
#pragma unroll
            for (int v = 0; v < 2; ++v) {
                const int jj = 4 * kc + ja + v;
                av[kc * 2 + v] = (jj == j0) ? (1.0f - fh) : ((jj == j0 + 1) ? fh : 0.0f);
            }
        a0[0] = av[0]; a0[1] = av[1];
        a1[0] = av[2]; a1[1] = av[3];
    }

    // B layout (fp32 4x16 per chunk): lane L, VGPR v holds T[4*kc + 2*(L/16) + v][L%16]
    // T[j][n] = sum_k G[j][k] * WW[w][k] = lerp over WW's two nonzeros
    const int wB = w0 + col;
    const float pw = (float)wB * (7.0f / 255.0f);
    int k0 = (int)pw; if (k0 > 6) k0 = 6;
    const float fw = pw - (float)k0;

    v8f f[3];
#pragma unroll
    for (int c = 0; c < 3; ++c) {
        const int r0 = lh << 1;
        v2f b0, b1;
        b0[0] = lerpf(Gs[c][r0 + 0][k0], Gs[c][r0 + 0][k0 + 1], fw);
        b0[1] = lerpf(Gs[c][r0 + 1][k0], Gs[c][r0 + 1][k0 + 1], fw);
        b1[0] = lerpf(Gs[c][r0 + 4][k0], Gs[c][r0 + 4][k0 + 1], fw);
        b1[1] = lerpf(Gs[c][r0 + 5][k0], Gs[c][r0 + 5][k0 + 1], fw);
        v8f acc = {0.f, 0.f, 0.f, 0.f, 0.f, 0.f, 0.f, 0.f};
        acc = __builtin_amdgcn_wmma_f32_16x16x4_f32(false, a0, false, b0,
                                                    (short)0, acc, false, false);
        acc = __builtin_amdgcn_wmma_f32_16x16x4_f32(false, a1, false, b1,
                                                    (short)0, acc, false, false);
        f[c] = acc;
    }

    // ---------- Stage 3: warp + trilinear gather (D layout: VGPR r -> h0 + r + 8*lh) ----------
    const float* __restrict__ xb = x   + (size_t)b * (64 * 256 * 256);
    float* __restrict__       ob = out + (size_t)b * (64 * 256 * 256);
    const int w = w0 + col;

#pragma unroll
    for (int r = 0; r < 8; ++r) {
        const int h = h0 + r + (lh << 3);
        // grid = base + flow; channel 0 -> x, 1 -> y, 2 -> z (faithful to reference)
        const float ix = reflectf((float)w + f[0][r] * 127.5f, 510.0f, 1.0f / 510.0f);
        const float iy = reflectf((float)h + f[1][r] * 127.5f, 510.0f, 1.0f / 510.0f);
        const float iz = reflectf((float)d + f[2][r] * 31.5f,  126.0f, 1.0f / 126.0f);

        int x0i = (int)ix; if (x0i > 254) x0i = 254;
        int y0i = (int)iy; if (y0i > 254) y0i = 254;
        int z0i = (int)iz; if (z0i > 62)  z0i = 62;
        const float fx = ix - (float)x0i;
        const float fy = iy - (float)y0i;
        const float fz = iz - (float)z0i;

        const int base = (z0i * 256 + y0i) * 256 + x0i;
        const float c000 = xb[base];
        const float c001 = xb[base + 1];
        const float c010 = xb[base + 256];
        const float c011 = xb[base + 257];
        const float c100 = xb[base + 65536];
        const float c101 = xb[base + 65537];
        const float c110 = xb[base + 65792];
        const float c111 = xb[base + 65793];

        const float c00 = lerpf(c000, c001, fx);
        const float c01 = lerpf(c010, c011, fx);
        const float c10 = lerpf(c100, c101, fx);
        const float c11 = lerpf(c110, c111, fx);
        const float c0  = lerpf(c00, c01, fy);
        const float c1  = lerpf(c10, c11, fy);
        ob[(d * 256 + h) * 256 + w] = lerpf(c0, c1, fz);
    }
}

extern "C" void kernel_launch(void* const* d_in, const int* in_sizes, int n_in,
                              void* d_out, int out_size, void* d_ws, size_t ws_size,
                              hipStream_t stream) {
    const float* x  = (const float*)d_in[0];   // (4,1,64,256,256) fp32
    const float* fc = (const float*)d_in[1];   // (4,3,8,8,8) fp32
    float* out = (float*)d_out;                // (4,1,64,256,256) fp32
    (void)in_sizes; (void)n_in; (void)out_size; (void)d_ws; (void)ws_size;
    elastic_warp_kernel<<<dim3(8192), dim3(256), 0, stream>>>(x, fc, out);
}